// GRAPH_AUTOENCODER_26465588478424
// MI455X (gfx1250) — compile-verified
//
#include <hip/hip_runtime.h>
#include <math.h>

typedef float v2f __attribute__((ext_vector_type(2)));
typedef float v8f __attribute__((ext_vector_type(8)));

// ---------------------------------------------------------------------------
// GEMM via V_WMMA_F32_16X16X4_F32 (full fp32 on the matrix pipe).
// Y[N,M] = act(X[N,K] @ W[K,M] + bias).  One wave computes a 32x32 tile
// (2 A-frags x 2 B-frags -> 4 WMMAs per K-step of 4).  Block = 4 waves =
// 128 rows x 32 cols.  All loads are unconditional (pointers clamped in
// range, out-of-range contributions zeroed by mask multiplies) so the hot
// loop has no EXEC manipulation.  K-tail (K % 4 != 0) peeled after the loop.
// act: 0=none, 1=relu, 2=sigmoid.
// ---------------------------------------------------------------------------
__global__ void k_gemm_wmma(const float* __restrict__ X, const float* __restrict__ W,
                            const float* __restrict__ bias, float* __restrict__ Y,
                            int N, int K, int M, int act)
{
    const int lane  = threadIdx.x & 31;
    const int wave  = threadIdx.x >> 5;
    const int mrow  = lane & 15;
    const int khalf = (lane >> 4) << 1;      // 0 for lanes 0-15, 2 for 16-31
    const int r0    = (blockIdx.x * 4 + wave) * 32;
    const int c0    = blockIdx.y * 32;

    const int ra0 = r0 + mrow,      ra1 = r0 + 16 + mrow;
    const int cb0 = c0 + mrow,      cb1 = c0 + 16 + mrow;
    const float am0 = (ra0 < N) ? 1.f : 0.f;
    const float am1 = (ra1 < N) ? 1.f : 0.f;
    const float bm0 = (cb0 < M) ? 1.f : 0.f;
    const float bm1 = (cb1 < M) ? 1.f : 0.f;
    const float* Xr0 = X + (size_t)(ra0 < N ? ra0 : N - 1) * K;
    const float* Xr1 = X + (size_t)(ra1 < N ? ra1 : N - 1) * K;
    const float* Wc0 = W + (cb0 < M ? cb0 : M - 1);
    const float* Wc1 = W + (cb1 < M ? cb1 : M - 1);

    v8f c00 = {0.f,0.f,0.f,0.f,0.f,0.f,0.f,0.f};
    v8f c01 = {0.f,0.f,0.f,0.f,0.f,0.f,0.f,0.f};
    v8f c10 = {0.f,0.f,0.f,0.f,0.f,0.f,0.f,0.f};
    v8f c11 = {0.f,0.f,0.f,0.f,0.f,0.f,0.f,0.f};

    const int K4 = K & ~3;
    for (int k0 = 0; k0 < K4; k0 += 4) {
        const int ka = k0 + khalf;
        __builtin_prefetch((const void*)(Xr0 + ka + 32), 0, 0);
        v2f a0 = *(const v2f*)(Xr0 + ka);          // contiguous (k, k+1)
        v2f a1 = *(const v2f*)(Xr1 + ka);
        v2f b0, b1;
        b0.x = Wc0[(size_t)ka * M];
        b0.y = Wc0[(size_t)(ka + 1) * M];
        b1.x = Wc1[(size_t)ka * M];
        b1.y = Wc1[(size_t)(ka + 1) * M];
        a0.x *= am0; a0.y *= am0;
        a1.x *= am1; a1.y *= am1;
        b0.x *= bm0; b0.y *= bm0;
        b1.x *= bm1; b1.y *= bm1;
        c00 = __builtin_amdgcn_wmma_f32_16x16x4_f32(false, a0, false, b0, (short)0, c00, false, false);
        c01 = __builtin_amdgcn_wmma_f32_16x16x4_f32(false, a0, false, b1, (short)0, c01, false, false);
        c10 = __builtin_amdgcn_wmma_f32_16x16x4_f32(false, a1, false, b0, (short)0, c10, false, false);
        c11 = __builtin_amdgcn_wmma_f32_16x16x4_f32(false, a1, false, b1, (short)0, c11, false, false);
    }
    if (K4 < K) {                                  // K tail (e.g. K=38)
        const int ka  = K4 + khalf;
        const int kc0 = (ka < K) ? ka : (K - 1);
        const int kc1 = (ka + 1 < K) ? (ka + 1) : (K - 1);
        const float k0m = (ka < K) ? 1.f : 0.f;
        const float k1m = (ka + 1 < K) ? 1.f : 0.f;
        v2f a0, a1, b0, b1;
        a0.x = Xr0[kc0] * (k0m * am0);  a0.y = Xr0[kc1] * (k1m * am0);
        a1.x = Xr1[kc0] * (k0m * am1);  a1.y = Xr1[kc1] * (k1m * am1);
        b0.x = Wc0[(size_t)kc0 * M] * (k0m * bm0);
        b0.y = Wc0[(size_t)kc1 * M] * (k1m * bm0);
        b1.x = Wc1[(size_t)kc0 * M] * (k0m * bm1);
        b1.y = Wc1[(size_t)kc1 * M] * (k1m * bm1);
        c00 = __builtin_amdgcn_wmma_f32_16x16x4_f32(false, a0, false, b0, (short)0, c00, false, false);
        c01 = __builtin_amdgcn_wmma_f32_16x16x4_f32(false, a0, false, b1, (short)0, c01, false, false);
        c10 = __builtin_amdgcn_wmma_f32_16x16x4_f32(false, a1, false, b0, (short)0, c10, false, false);
        c11 = __builtin_amdgcn_wmma_f32_16x16x4_f32(false, a1, false, b1, (short)0, c11, false, false);
    }

    const float bv0 = (bias != nullptr && cb0 < M) ? bias[cb0] : 0.f;
    const float bv1 = (bias != nullptr && cb1 < M) ? bias[cb1] : 0.f;

    auto store_tile = [&](const v8f& c, int rbase, int col, float bv) {
        #pragma unroll
        for (int v = 0; v < 8; ++v) {
            const int rr = rbase + v + ((lane >> 4) << 3);
            if (rr < N && col < M) {
                float val = c[v] + bv;
                if (act == 1)      val = fmaxf(val, 0.f);
                else if (act == 2) val = 1.f / (1.f + __expf(-val));
                Y[(size_t)rr * M + col] = val;
            }
        }
    };
    store_tile(c00, r0,      cb0, bv0);
    store_tile(c01, r0,      cb1, bv1);
    store_tile(c10, r0 + 16, cb0, bv0);
    store_tile(c11, r0 + 16, cb1, bv1);
}

// ---------------------------------------------------------------------------
// Elementwise / scatter helpers
// ---------------------------------------------------------------------------
__global__ void k_fill(float* p, long long n, float v)
{
    long long i = (long long)blockIdx.x * 256 + threadIdx.x;
    const long long s = (long long)gridDim.x * 256;
    for (; i < n; i += s) p[i] = v;
}

__global__ void k_degree(const int* __restrict__ dst, float* deg, int E)
{
    int e = blockIdx.x * 256 + threadIdx.x;
    if (e < E) atomicAdd(&deg[dst[e]], 1.f);
}

__global__ void k_rsqrt(const float* __restrict__ deg, float* dinv, int N)
{
    int i = blockIdx.x * 256 + threadIdx.x;
    if (i < N) dinv[i] = rsqrtf(deg[i]);
}

// agg[i,f] = dinv[i]^2 * g[i,f] + bias[f]      (self-loop term + bias)
__global__ void k_self(const float* __restrict__ g, const float* __restrict__ dinv,
                       const float* __restrict__ bias, float* __restrict__ agg,
                       int N, int M)
{
    const int i = blockIdx.x;
    const int f = threadIdx.x;
    const float di = dinv[i];
    agg[(size_t)i * M + f] = di * di * g[(size_t)i * M + f]
                           + (bias ? bias[f] : 0.f);
}

// agg[dst,f] += dinv[src]*dinv[dst] * g[src,f]
__global__ void k_edge_scatter(const int* __restrict__ src, const int* __restrict__ dst,
                               const float* __restrict__ dinv, const float* __restrict__ g,
                               float* agg, int E, int M, int mShift)
{
    const int t = blockIdx.x * 256 + threadIdx.x;
    const int e = t >> mShift;
    const int f = t & (M - 1);
    if (e >= E) return;
    const int s = src[e], d = dst[e];
    const float coef = dinv[s] * dinv[d];
    atomicAdd(&agg[(size_t)d * M + f], coef * g[(size_t)s * M + f]);
}

// in-place relu + per-column sum / sumsq accumulation (M = 128 or 256)
__global__ void k_relu_colstats(float* x, long long total, int M,
                                float* colsum, float* colsq)
{
    const int t = threadIdx.x;
    const int col = t & (M - 1);
    float s = 0.f, q = 0.f;
    const long long stride = (long long)gridDim.x * 256;
    for (long long i = (long long)blockIdx.x * 256 + t; i < total; i += stride) {
        const float v = fmaxf(x[i], 0.f);
        x[i] = v;
        s += v;
        q += v * v;
    }
    __shared__ float ss[256], sq[256];
    ss[t] = s; sq[t] = q;
    __syncthreads();
    if (t < M) {
        for (int j = t + M; j < 256; j += M) { s += ss[j]; q += sq[j]; }
        atomicAdd(&colsum[col], s);
        atomicAdd(&colsq[col],  q);
    }
}

__global__ void k_bn_apply(float* x, long long total, int M, float invN,
                           const float* __restrict__ colsum,
                           const float* __restrict__ colsq)
{
    long long i = (long long)blockIdx.x * 256 + threadIdx.x;
    const long long stride = (long long)gridDim.x * 256;
    for (; i < total; i += stride) {
        const int c = (int)(i & (M - 1));
        const float m = colsum[c] * invN;
        const float v = colsq[c] * invN - m * m;
        x[i] = (x[i] - m) * rsqrtf(v + 1e-5f);
    }
}

// per-row L2 normalize, F=64, one wave32 per row
__global__ void k_l2norm(float* x, int N)
{
    const int lane = threadIdx.x & 31;
    const int wave = threadIdx.x >> 5;
    const int row  = blockIdx.x * 8 + wave;
    if (row >= N) return;
    const float a = x[(size_t)row * 64 + lane];
    const float b = x[(size_t)row * 64 + 32 + lane];
    float s = a * a + b * b;
    for (int m = 16; m > 0; m >>= 1) s += __shfl_xor(s, m, 32);
    const float inv = 1.f / fmaxf(sqrtf(s), 1e-12f);
    x[(size_t)row * 64 + lane]      = a * inv;
    x[(size_t)row * 64 + 32 + lane] = b * inv;
}

__global__ void k_seg_cnt(const int* __restrict__ seg, float* cnt, int N)
{
    int i = blockIdx.x * 256 + threadIdx.x;
    if (i < N) atomicAdd(&cnt[seg[i]], 1.f);
}

// F = 64 pooling sum
__global__ void k_seg_sum(const float* __restrict__ x, const int* __restrict__ seg,
                          float* sum, int N)
{
    const int i = blockIdx.x;
    const int f = threadIdx.x;
    atomicAdd(&sum[(size_t)seg[i] * 64 + f], x[(size_t)i * 64 + f]);
}

__global__ void k_seg_div(float* pool, const float* __restrict__ cnt, int G)
{
    const int i = blockIdx.x * 256 + threadIdx.x;
    if (i < G * 64) pool[i] /= fmaxf(cnt[i >> 6], 1.f);
}

// in-place relu + per-segment sum / sumsq
__global__ void k_seg_stats(float* x, const int* __restrict__ seg,
                            float* ssum, float* ssq, int M)
{
    const int i = blockIdx.x;
    const int f = threadIdx.x;
    const float v = fmaxf(x[(size_t)i * M + f], 0.f);
    x[(size_t)i * M + f] = v;
    const int s = seg[i];
    atomicAdd(&ssum[(size_t)s * M + f], v);
    atomicAdd(&ssq[(size_t)s * M + f],  v * v);
}

__global__ void k_seg_bn_apply(float* x, const int* __restrict__ seg,
                               const float* __restrict__ cnt,
                               const float* __restrict__ ssum,
                               const float* __restrict__ ssq, int M)
{
    const int i = blockIdx.x;
    const int f = threadIdx.x;
    const int s = seg[i];
    const float c = fmaxf(cnt[s], 1.f);
    const float m = ssum[(size_t)s * M + f] / c;
    const float v = ssq[(size_t)s * M + f] / c - m * m;
    x[(size_t)i * M + f] = (x[(size_t)i * M + f] - m) * rsqrtf(fmaxf(v, 0.f) + 1e-5f);
}

// ---------------------------------------------------------------------------
extern "C" void kernel_launch(void* const* d_in, const int* in_sizes, int n_in,
                              void* d_out, int out_size, void* d_ws, size_t ws_size,
                              hipStream_t stream)
{
    const int NN = 200000, NE = 800000, NG = 2048, NS = 32768, ES = 131072;
    const int FI = 38, H0 = 256, H1 = 128, H2 = 64;

    const float* x        = (const float*)d_in[0];
    const int*   ei       = (const int*)d_in[1];
    const int*   batch    = (const int*)d_in[2];
    const float* target_x = (const float*)d_in[3];
    const float* pos_x    = (const float*)d_in[4];
    const int*   pos_ei   = (const int*)d_in[5];
    const int*   pos_b    = (const int*)d_in[6];
    const float* neg_x    = (const float*)d_in[7];
    const int*   neg_ei   = (const int*)d_in[8];
    const int*   neg_b    = (const int*)d_in[9];
    const float* eW0 = (const float*)d_in[10]; const float* eb0 = (const float*)d_in[11];
    const float* eW1 = (const float*)d_in[12]; const float* eb1 = (const float*)d_in[13];
    const float* eW2 = (const float*)d_in[14]; const float* eb2 = (const float*)d_in[15];
    const float* dW0 = (const float*)d_in[16];
    const float* dW1 = (const float*)d_in[17];
    const float* dW2 = (const float*)d_in[18];
    const float* nW0 = (const float*)d_in[19]; const float* nb0 = (const float*)d_in[20];
    const float* nW1 = (const float*)d_in[21]; const float* nb1 = (const float*)d_in[22];
    const float* nW2 = (const float*)d_in[23]; const float* nb2 = (const float*)d_in[24];
    const float* sW0 = (const float*)d_in[25]; const float* sb0 = (const float*)d_in[26];
    const float* sW1 = (const float*)d_in[27]; const float* sb1 = (const float*)d_in[28];
    const float* cW0 = (const float*)d_in[29]; const float* cb0 = (const float*)d_in[30];
    const float* cW1 = (const float*)d_in[31]; const float* cb1 = (const float*)d_in[32];
    const float* cW2 = (const float*)d_in[33]; const float* cb2 = (const float*)d_in[34];
    (void)in_sizes; (void)n_in; (void)out_size; (void)ws_size;

    float* out      = (float*)d_out;
    float* x_recon  = out;                                // 200000*38
    float* z_pool   = out + (size_t)NN * FI;              // 2048*64
    float* target_z = z_pool + (size_t)NG * H2;
    float* pos_pool = target_z + (size_t)NG * H2;
    float* neg_pool = pos_pool + (size_t)NG * H2;
    float* neg_pred = neg_pool + (size_t)NG * H2;         // 2048

    float* ws = (float*)d_ws;
    const size_t BIG = (size_t)NN * H0;                   // 51.2M floats
    float* bufA = ws;
    float* bufB = ws + BIG;
    float* p = ws + 2 * BIG;
    float* deg    = p; p += NN;
    float* dinv   = p; p += NN;
    float* colsum = p; p += 256;
    float* colsq  = p; p += 256;
    float* segcnt = p; p += NG;
    float* segsum = p; p += (size_t)NG * 256;
    float* segsq  = p; p += (size_t)NG * 256;
    float* tbuf0  = p; p += (size_t)NG * 256;
    float* tbuf1  = p; p += (size_t)NG * 256;

    auto gemm = [&](const float* X, const float* W, const float* bias, float* Y,
                    int N, int K, int M, int act) {
        dim3 grid((N + 127) / 128, (M + 31) / 32);
        k_gemm_wmma<<<grid, dim3(128), 0, stream>>>(X, W, bias, Y, N, K, M, act);
    };
    auto fill = [&](float* q, long long n, float v) {
        int blocks = (int)((n + 255) / 256); if (blocks > 4096) blocks = 4096;
        k_fill<<<dim3(blocks), dim3(256), 0, stream>>>(q, n, v);
    };
    auto degree = [&](const int* d, int N, int E) {
        fill(deg, N, 1.f);
        k_degree<<<dim3((E + 255) / 256), dim3(256), 0, stream>>>(d, deg, E);
        k_rsqrt<<<dim3((N + 255) / 256), dim3(256), 0, stream>>>(deg, dinv, N);
    };
    auto gcn_agg = [&](const float* g, float* agg, const float* bias,
                       const int* s, const int* d, int N, int E, int M) {
        k_self<<<dim3(N), dim3(M), 0, stream>>>(g, dinv, bias, agg, N, M);
        const int mShift = (M == 256) ? 8 : (M == 128) ? 7 : 6;
        const long long tt = (long long)E * M;
        k_edge_scatter<<<dim3((unsigned)((tt + 255) / 256)), dim3(256), 0, stream>>>(
            s, d, dinv, g, agg, E, M, mShift);
    };
    auto bn = [&](float* X, int N, int M) {
        fill(colsum, 256, 0.f); fill(colsq, 256, 0.f);
        k_relu_colstats<<<dim3(1024), dim3(256), 0, stream>>>(X, (long long)N * M, M, colsum, colsq);
        k_bn_apply<<<dim3(2048), dim3(256), 0, stream>>>(X, (long long)N * M, M, 1.f / (float)N, colsum, colsq);
    };
    auto seg_mean = [&](const float* z, const int* seg, float* pool, int N) {
        fill(segcnt, NG, 0.f); fill(pool, (long long)NG * 64, 0.f);
        k_seg_cnt<<<dim3((N + 255) / 256), dim3(256), 0, stream>>>(seg, segcnt, N);
        k_seg_sum<<<dim3(N), dim3(64), 0, stream>>>(z, seg, pool, N);
        k_seg_div<<<dim3((NG * 64 + 255) / 256), dim3(256), 0, stream>>>(pool, segcnt, NG);
    };
    auto seg_bn = [&](float* X, const int* seg, int N, int M) {
        fill(segcnt, NG, 0.f);
        fill(segsum, (long long)NG * M, 0.f); fill(segsq, (long long)NG * M, 0.f);
        k_seg_cnt<<<dim3((N + 255) / 256), dim3(256), 0, stream>>>(seg, segcnt, N);
        k_seg_stats<<<dim3(N), dim3(M), 0, stream>>>(X, seg, segsum, segsq, M);
        k_seg_bn_apply<<<dim3(N), dim3(M), 0, stream>>>(X, seg, segcnt, segsum, segsq, M);
    };

    const int* src = ei;
    const int* dst = ei + NE;

    // ---------------- main GCN encoder ----------------
    degree(dst, NN, NE);

    // L0: 38 -> 256
    gemm(x, eW0, nullptr, bufA, NN, FI, H0, 0);
    gcn_agg(bufA, bufB, eb0, src, dst, NN, NE, H0);
    bn(bufB, NN, H0);                                     // h0 = bufB

    // L1: 256 -> 128
    float* g1 = bufA;
    float* h1 = bufA + BIG / 2;
    gemm(bufB, eW1, nullptr, g1, NN, H0, H1, 0);
    gcn_agg(g1, h1, eb1, src, dst, NN, NE, H1);
    bn(h1, NN, H1);

    // L2: 128 -> 64, then L2-normalize
    float* g2 = bufB;
    float* z  = bufA;
    gemm(h1, eW2, nullptr, g2, NN, H1, H2, 0);
    gcn_agg(g2, z, eb2, src, dst, NN, NE, H2);
    k_l2norm<<<dim3((NN + 7) / 8), dim3(256), 0, stream>>>(z, NN);

    // z_pool
    seg_mean(z, batch, z_pool, NN);

    // ---------------- decoder ----------------
    float* r0 = bufB;
    gemm(z, dW0, nullptr, r0, NN, H2, H1, 1);             // relu(z @ dec_W0)
    float* r1 = bufA;                                     // z dead now
    gemm(r0, dW1, nullptr, r1, NN, H1, H0, 1);            // relu
    gemm(r1, dW2, nullptr, x_recon, NN, H0, FI, 2);       // sigmoid

    // ---------------- target node MLP ----------------
    gemm(target_x, nW0, nb0, tbuf0, NG, FI, H0, 1);
    bn(tbuf0, NG, H0);
    gemm(tbuf0, nW1, nb1, tbuf1, NG, H0, H1, 1);
    bn(tbuf1, NG, H1);
    gemm(tbuf1, nW2, nb2, target_z, NG, H1, H2, 0);
    k_l2norm<<<dim3((NG + 7) / 8), dim3(256), 0, stream>>>(target_z, NG);

    // ---------------- subgraph encoder (pos / neg) ----------------
    auto encode_sub = [&](const float* sx, const int* sei, const int* sb, float* pool) {
        const int* ss = sei;
        const int* sd = sei + ES;
        degree(sd, NS, ES);
        float* sA = bufB;
        float* sB = bufB + (size_t)NS * H0;
        // layer 0: 38 -> 256
        gemm(sx, sW0, nullptr, sA, NS, FI, H0, 0);
        gcn_agg(sA, sB, sb0, ss, sd, NS, ES, H0);
        seg_bn(sB, sb, NS, H0);
        // layer 1: 256 -> 128
        gemm(sB, sW1, nullptr, sA, NS, H0, H1, 0);
        gcn_agg(sA, sB, sb1, ss, sd, NS, ES, H1);
        seg_bn(sB, sb, NS, H1);
        // layer 2 reuses main encoder last layer: 128 -> 64
        gemm(sB, eW2, nullptr, sA, NS, H1, H2, 0);
        gcn_agg(sA, sB, eb2, ss, sd, NS, ES, H2);
        k_l2norm<<<dim3((NS + 7) / 8), dim3(256), 0, stream>>>(sB, NS);
        seg_mean(sB, sb, pool, NS);
    };
    encode_sub(pos_x, pos_ei, pos_b, pos_pool);
    encode_sub(neg_x, neg_ei, neg_b, neg_pool);

    // ---------------- classifier on neg_pool ----------------
    gemm(neg_pool, cW0, cb0, tbuf0, NG, H2, H1, 1);
    gemm(tbuf0, cW1, cb1, tbuf1, NG, H1, 16, 1);
    gemm(tbuf1, cW2, cb2, neg_pred, NG, 16, 1, 2);
}